// CTProjector_38946763440496
// MI455X (gfx1250) — compile-verified
//
#include <hip/hip_runtime.h>
#include <cmath>

typedef __attribute__((ext_vector_type(2))) float v2f;
typedef __attribute__((ext_vector_type(8))) float v8f;

#define WPB   8              // waves per block (wave32)
#define BLOCK (WPB * 32)

#if __has_builtin(__builtin_amdgcn_wmma_f32_16x16x4_f32)
#define HAVE_WMMA4 1
#else
#define HAVE_WMMA4 0
#endif

__device__ __forceinline__ float safe_inv(float d) {
  float s = (fabsf(d) < 1e-9f) ? ((d < 0.0f) ? -1e-9f : 1e-9f) : d;
  return 1.0f / s;
}
__device__ __forceinline__ int iclampi(int v, int lo, int hi) {
  return v < lo ? lo : (v > hi ? hi : v);
}
__device__ __forceinline__ float fclampf(float v, float lo, float hi) {
  return fminf(fmaxf(v, lo), hi);
}
// WGP-scope prefetch (scope field omitted -> SCOPE_WGP): pulls into ALL cache
// levels on miss, unlike the builtin's SE/DEV lowering which stops at GL2.
__device__ __forceinline__ void prefetch_wgp(const float* p) {
  asm volatile("global_prefetch_b8 %0, off" :: "v"(p) : "memory");
}

__global__ __launch_bounds__(BLOCK) void ct_fwd_wmma(
    const float* __restrict__ vol,
    const float* __restrict__ sources,   // Ns x 3
    const float* __restrict__ dests,     // Nd x 3
    const float* __restrict__ vstart,    // 3
    const float* __restrict__ vspace,    // 3
    const int*   __restrict__ nsteps_p,  // 1
    float* __restrict__ partial,         // splitCount x NRAYS
    int Ns, int Nd, int D, int H, int W,
    int numGroups, int splitCount)
{
  const int NRAYS = Ns * Nd;
  // force wave id into an SGPR so loop bounds & group/split are provably uniform
  const int wave = __builtin_amdgcn_readfirstlane(blockIdx.x * WPB + (threadIdx.x >> 5));
  if (wave >= numGroups * splitCount) return;   // uniform: EXEC stays all-ones
  const int group = wave % numGroups;
  const int split = wave / numGroups;
  const int lane  = threadIdx.x & 31;
  const int m     = lane & 15;     // ray row within the 16-ray group
  const int khalf = lane >> 4;     // 0: steps k+0/k+1, 1: steps k+2/k+3 (A-matrix layout)

  const int  ray   = group * 16 + m;
  const bool rayOK = (ray < NRAYS);
  const int  r     = rayOK ? ray : 0;
  const int  sIdx  = r / Nd;
  const int  dIdx  = r - sIdx * Nd;

  const float sx = sources[sIdx * 3 + 0], sy = sources[sIdx * 3 + 1], sz = sources[sIdx * 3 + 2];
  const float ex = dests[dIdx * 3 + 0],  ey = dests[dIdx * 3 + 1],  ez = dests[dIdx * 3 + 2];
  const float v0 = vstart[0], v1 = vstart[1], v2 = vstart[2];
  const float p0 = vspace[0], p1 = vspace[1], p2 = vspace[2];
  const int   nsteps = __builtin_amdgcn_readfirstlane(*nsteps_p);

  // ray direction + slab clip (matches reference exactly)
  const float dx = ex - sx, dy = ey - sy, dz = ez - sz;
  const float bx1 = v0 + p0 * (float)(D - 1);
  const float by1 = v1 + p1 * (float)(H - 1);
  const float bz1 = v2 + p2 * (float)(W - 1);
  const float ivx = safe_inv(dx), ivy = safe_inv(dy), ivz = safe_inv(dz);
  const float tax = (v0  - sx) * ivx, tbx = (bx1 - sx) * ivx;
  const float tay = (v1  - sy) * ivy, tby = (by1 - sy) * ivy;
  const float taz = (v2  - sz) * ivz, tbz = (bz1 - sz) * ivz;
  float tn = fmaxf(fmaxf(fminf(tax, tbx), fminf(tay, tby)), fminf(taz, tbz));
  float tf = fminf(fminf(fmaxf(tax, tbx), fmaxf(tay, tby)), fmaxf(taz, tbz));
  tn = fclampf(tn, 0.0f, 1.0f);
  tf = fclampf(tf, 0.0f, 1.0f);
  const bool  valid = tf > tn;
  const float rlen  = sqrtf(dx * dx + dy * dy + dz * dz);
  const float tstep = (tf - tn) / (float)nsteps;
  const float wgt   = (valid && rayOK) ? (rlen * tstep) : 0.0f;  // dt, zeroed if inactive

  // grid-space ray: g(t) = og + t * rg  (hoists the per-sample subtract/divide)
  const float ogx = (sx - v0) / p0, rgx = dx / p0;
  const float ogy = (sy - v1) / p1, rgy = dy / p1;
  const float ogz = (sz - v2) / p2, rgz = dz / p2;
  const float gmx = (float)(D - 1), gmy = (float)(H - 1), gmz = (float)(W - 1);
  const unsigned Wu  = (unsigned)W;            // 32-bit element-index math:
  const unsigned sXu = (unsigned)(H * W);      // volume < 2^31 elements

  const int lo = __builtin_amdgcn_readfirstlane((int)(((long long)nsteps * split)       / splitCount));
  const int hi = __builtin_amdgcn_readfirstlane((int)(((long long)nsteps * (split + 1)) / splitCount));

  // one sample: trilinear + bounds, pre-scaled by dt (branchless -> EXEC stays full)
  auto sample = [&](int i, bool ok) -> float {
    const float t  = tn + ((float)i + 0.5f) * tstep;
    const float gx = ogx + t * rgx;
    const float gy = ogy + t * rgy;
    const float gz = ogz + t * rgz;
    const int x0 = iclampi((int)floorf(gx), 0, D - 2);
    const int y0 = iclampi((int)floorf(gy), 0, H - 2);
    const int z0 = iclampi((int)floorf(gz), 0, W - 2);
    const float fx = fclampf(gx - (float)x0, 0.0f, 1.0f);
    const float fy = fclampf(gy - (float)y0, 0.0f, 1.0f);
    const float fz = fclampf(gz - (float)z0, 0.0f, 1.0f);
    const unsigned idx = ((unsigned)x0 * sXu) + ((unsigned)y0 * Wu) + (unsigned)z0;
    const float c000 = vol[idx],            c001 = vol[idx + 1u];
    const float c010 = vol[idx + Wu],       c011 = vol[idx + Wu + 1u];
    const float c100 = vol[idx + sXu],      c101 = vol[idx + sXu + 1u];
    const float c110 = vol[idx + sXu + Wu], c111 = vol[idx + sXu + Wu + 1u];
    const float omz = 1.0f - fz, omy = 1.0f - fy, omx = 1.0f - fx;
    const float c00 = c000 * omz + c001 * fz;
    const float c01 = c010 * omz + c011 * fz;
    const float c10 = c100 * omz + c101 * fz;
    const float c11 = c110 * omz + c111 * fz;
    const float c0  = c00 * omy + c01 * fy;
    const float c1  = c10 * omy + c11 * fy;
    const float val = c0 * omx + c1 * fx;
    const bool inb = (gx >= 0.0f) & (gx <= gmx) &
                     (gy >= 0.0f) & (gy <= gmy) &
                     (gz >= 0.0f) & (gz <= gmz);
    return (inb & ok) ? (wgt * val) : 0.0f;
  };

  v8f acc = {0.f, 0.f, 0.f, 0.f, 0.f, 0.f, 0.f, 0.f};
  v2f ones; ones.x = 1.0f; ones.y = 1.0f;   // B all-ones => WMMA = row-sum over K
#if !HAVE_WMMA4
  float facc = 0.0f;
#endif

  // 8 steps (2 WMMA chunks) per iteration: ~32 gathers in flight per lane
  for (int base = lo; base < hi; base += 8) {
    const int s0 = base + khalf * 2;       // chunk 0 steps (this lane)
    const int s1 = s0 + 4;                 // chunk 1 steps (this lane)

    // prefetch next iteration's first sample into the WGP cache (address clamped in-bounds)
    {
      const float t  = tn + ((float)(s0 + 8) + 0.5f) * tstep;
      const float gx = ogx + t * rgx, gy = ogy + t * rgy, gz = ogz + t * rgz;
      const int px = iclampi((int)floorf(gx), 0, D - 2);
      const int py = iclampi((int)floorf(gy), 0, H - 2);
      const int pz = iclampi((int)floorf(gz), 0, W - 2);
      prefetch_wgp(vol + ((unsigned)px * sXu + (unsigned)py * Wu + (unsigned)pz));
    }

    v2f a0, a1;
    a0.x = sample(s0,     s0     < hi);
    a0.y = sample(s0 + 1, s0 + 1 < hi);
    a1.x = sample(s1,     s1     < hi);
    a1.y = sample(s1 + 1, s1 + 1 < hi);
#if HAVE_WMMA4
    // D[m][*] += sum_k A[m][k] : running per-ray line-integral in the accumulator
    acc = __builtin_amdgcn_wmma_f32_16x16x4_f32(false, a0, false, ones,
                                                (short)0, acc, false, false);
    acc = __builtin_amdgcn_wmma_f32_16x16x4_f32(false, a1, false, ones,
                                                (short)0, acc, false, false);
#else
    facc += (a0.x + a0.y) + (a1.x + a1.y);
#endif
  }

#if HAVE_WMMA4
  // C/D layout: VGPR j, lanes 0-15 -> M=j ; lanes 16-31 -> M=8+j (any column N)
  if (m == 0) {  // lanes 0 and 16 export rows 0-7 / 8-15
    const int rbase = group * 16 + khalf * 8;
#pragma unroll
    for (int j = 0; j < 8; ++j) {
      const int rr = rbase + j;
      if (rr < NRAYS) partial[(long long)split * NRAYS + rr] = acc[j];
    }
  }
#else
  float tot = facc + __shfl_xor(facc, 16, 32);
  if (lane < 16 && rayOK) partial[(long long)split * NRAYS + ray] = tot;
#endif
}

__global__ void ct_reduce(const float* __restrict__ partial,
                          float* __restrict__ out, int NRAYS, int splitCount)
{
  const int i = blockIdx.x * blockDim.x + threadIdx.x;
  if (i >= NRAYS) return;
  float s = 0.0f;
  for (int k = 0; k < splitCount; ++k)
    s += partial[(long long)k * NRAYS + i];
  out[i] = s;
}

extern "C" void kernel_launch(void* const* d_in, const int* in_sizes, int n_in,
                              void* d_out, int out_size, void* d_ws, size_t ws_size,
                              hipStream_t stream) {
  const float* vol     = (const float*)d_in[0];
  const float* sources = (const float*)d_in[1];
  const float* dests   = (const float*)d_in[2];
  const float* vstart  = (const float*)d_in[3];
  const float* vspace  = (const float*)d_in[4];
  const int*   nsteps  = (const int*)d_in[5];

  const int Ns    = in_sizes[1] / 3;
  const int Nd    = in_sizes[2] / 3;
  const int NRAYS = Ns * Nd;
  const int D = (int)llround(cbrt((double)in_sizes[0]));  // cubic volume (256)
  const int H = D, W = D;
  const int numGroups = (NRAYS + 15) / 16;

  int splitCount = 8;                                   // 8x step-split -> 8192 waves
  size_t need = (size_t)splitCount * (size_t)NRAYS * sizeof(float);
  float* partial = (float*)d_ws;
  const bool haveWs = (d_ws != nullptr) && (ws_size >= need);
  if (!haveWs) { splitCount = 1; partial = (float*)d_out; }

  const int waves  = numGroups * splitCount;
  const int blocks = (waves + WPB - 1) / WPB;
  ct_fwd_wmma<<<blocks, BLOCK, 0, stream>>>(vol, sources, dests, vstart, vspace,
                                            nsteps, partial, Ns, Nd, D, H, W,
                                            numGroups, splitCount);
  if (haveWs) {
    const int rb = (NRAYS + 255) / 256;
    ct_reduce<<<rb, 256, 0, stream>>>(partial, (float*)d_out, NRAYS, splitCount);
  }
}